// crossCorrelation3D_53738630807798
// MI455X (gfx1250) — compile-verified
//
#include <hip/hip_runtime.h>

typedef float v2f __attribute__((ext_vector_type(2)));
typedef float v8f __attribute__((ext_vector_type(8)));

#define BD 2
#define DD 160
#define HD 192
#define WD 160
#define RAD 4
#define PATCH 24            // 16 + 2*RAD
#define PATCH_P 26          // padded row stride (keeps 8B alignment, no conflicts)
#define YB_P 26
#define CD 20               // z outputs per block
#define NPLANES (CD + 2*RAD)
#define RING_P 20           // ring row stride (16B-aligned float4 slots)

static __device__ inline v8f wmma_fp32(v2f a, v2f b, v8f c) {
  // D = A(16x4,f32) * B(4x16,f32) + C(16x16,f32)
  return __builtin_amdgcn_wmma_f32_16x16x4_f32(
      /*neg_a=*/false, a, /*neg_b=*/false, b,
      /*c_mod=*/(short)0, c, /*reuse_a=*/false, /*reuse_b=*/false);
}

__global__ __launch_bounds__(32)
void ncc3d_partial(const float* __restrict__ I, const float* __restrict__ T,
                   float* __restrict__ partials)
{
  // patchT[s][w][h]: transposed halo patches so a lane's two K-values are a
  // contiguous, 8B-aligned pair -> ds_load_b64 straight into WMMA operands.
  __shared__ __align__(16) float patchT[5][PATCH][PATCH_P];
  __shared__ __align__(16) float ybuf[2][16][YB_P];       // ping-pong Y(16x24)
  __shared__ __align__(16) float ringT[5][9][16][RING_P]; // [col][row]: b128able

  const int tid  = threadIdx.x;   // 0..31, one wave
  const int half = tid >> 4;      // 0 or 1
  const int l16  = tid & 15;

  const int tilesW = WD / 16;                 // 10
  const int tw = blockIdx.x % tilesW;
  const int th = blockIdx.x / tilesW;         // 0..11
  const int z0 = blockIdx.y * CD;
  const int b  = blockIdx.z;
  const int h0 = th * 16;
  const int w0 = tw * 16;

  // zero the ring buffer
  for (int i = tid; i < 5*9*16*RING_P; i += 32) (&ringT[0][0][0][0])[i] = 0.0f;
  __syncthreads();

  // Constant banded ones matrix chunks: band(m,k) = 1 iff k in [m, m+8].
  // Serves as A-operand (Bv, m=l16) in stage 1 and B-operand (Bt, n=l16)
  // in stage 2 (identical lane values by symmetry of the layouts).
  v2f bnd[6];
  for (int kt = 0; kt < 6; ++kt) {
    const int k = 4*kt + 2*half;
    bnd[kt].x = (k   >= l16 && k   <= l16 + 8) ? 1.0f : 0.0f;
    bnd[kt].y = (k+1 >= l16 && k+1 <= l16 + 8) ? 1.0f : 0.0f;
  }

  const v8f vzero = {0.f,0.f,0.f,0.f,0.f,0.f,0.f,0.f};
  v8f run[5];
  for (int s = 0; s < 5; ++s) run[s] = vzero;
  float acc = 0.0f;

  const long planeHW = (long)HD * WD;
  const float* Ib = I + (long)b * DD * planeHW;
  const float* Tb = T + (long)b * DD * planeHW;

  for (int it = 0; it < NPLANES; ++it) {
    const int d = z0 - RAD + it;
    const int slot = it % 9;
    v8f S[5];

    if (d >= 0 && d < DD) {
      __syncthreads();   // protect prior-plane LDS reads from overwrite
      // Load 24x24 halo patches of I and T (zero-padded), build products,
      // store transposed: patchT[s][w-idx][h-idx].
      for (int idx = tid; idx < PATCH*PATCH; idx += 32) {
        const int r = idx / PATCH, c = idx % PATCH;   // r: h-offset, c: w-offset
        const int h = h0 - RAD + r;
        const int w = w0 - RAD + c;
        float iv = 0.f, tv = 0.f;
        if (h >= 0 && h < HD && w >= 0 && w < WD) {
          const long off = (long)d * planeHW + (long)h * WD + w;
          iv = Ib[off];
          tv = Tb[off];
        }
        patchT[0][c][r] = iv;
        patchT[1][c][r] = tv;
        patchT[2][c][r] = iv*iv;
        patchT[3][c][r] = tv*tv;
        patchT[4][c][r] = iv*tv;
      }
      __syncthreads();

      // Per statistic: 2D box sum S = Bv(16x24) * X(24x24) * Bt(24x16)
      for (int s = 0; s < 5; ++s) {
        float (*yb)[YB_P] = ybuf[s & 1];
        // Stage 1: Y(16x24) = Bv * X, two overlapping 16-col N-tiles
        for (int t = 0; t < 2; ++t) {
          const int c0 = t * 8;       // cols 0..15 then 8..23
          v8f cacc = vzero;
          for (int kt = 0; kt < 6; ++kt) {
            const int k0 = 4*kt;
            // B operand rows k0..k0+3: lane gets X[k0+2*half .. +1][c0+l16],
            // contiguous in patchT -> single ds_load_b64
            const v2f bx =
                *reinterpret_cast<const v2f*>(&patchT[s][c0 + l16][k0 + 2*half]);
            cacc = wmma_fp32(bnd[kt], bx, cacc);
          }
          for (int v = 0; v < 8; ++v)          // C/D layout: row v + 8*half
            yb[v + 8*half][c0 + l16] = cacc[v];
        }
        __syncthreads();   // Y visible before stage-2 reads (ping-pong covers WAR)
        // Stage 2: S(16x16) = Y * Bt, K=24
        v8f sacc = vzero;
        for (int kt = 0; kt < 6; ++kt) {
          const int k0 = 4*kt;
          // A operand: Y[m=l16][k0+2*half .. +1], contiguous -> ds_load_b64
          const v2f ax = *reinterpret_cast<const v2f*>(&yb[l16][k0 + 2*half]);
          sacc = wmma_fp32(ax, bnd[kt], sacc);
        }
        S[s] = sacc;
      }
    } else {
      for (int s = 0; s < 5; ++s) S[s] = vzero;   // zero-padded z plane
    }

    // Depth running sum via 9-plane ring: run += S_new - S_old.
    // ringT[s][slot][col][row]: a lane's 8 rows are contiguous -> b128 ops.
    for (int s = 0; s < 5; ++s) {
      float4* rp0 = reinterpret_cast<float4*>(&ringT[s][slot][l16][8*half]);
      float4* rp1 = reinterpret_cast<float4*>(&ringT[s][slot][l16][8*half + 4]);
      const float4 o0 = *rp0;
      const float4 o1 = *rp1;
      v8f old;
      old[0] = o0.x; old[1] = o0.y; old[2] = o0.z; old[3] = o0.w;
      old[4] = o1.x; old[5] = o1.y; old[6] = o1.z; old[7] = o1.w;
      run[s] = run[s] + S[s] - old;
      const float4 n0 = {S[s][0], S[s][1], S[s][2], S[s][3]};
      const float4 n1 = {S[s][4], S[s][5], S[s][6], S[s][7]};
      *rp0 = n0;
      *rp1 = n1;
    }

    // Output plane z = d - RAD becomes valid once 9 planes accumulated
    if (it >= 2*RAD) {
      const float ks = 729.0f, inv = 1.0f / 729.0f;
      for (int v = 0; v < 8; ++v) {
        const float Is  = run[0][v], Ts  = run[1][v];
        const float IIs = run[2][v], TTs = run[3][v], ITs = run[4][v];
        const float Ihat = Is * inv, That = Ts * inv;
        const float cross = ITs - Ihat*Ts - That*Is + That*Ihat*ks;
        const float Tvar  = TTs - 2.0f*That*Ts + That*That*ks;
        const float Ivar  = IIs - 2.0f*Ihat*Is + Ihat*Ihat*ks;
        acc += (cross*cross) / (Tvar*Ivar + 1e-5f);
      }
    }
  }

  // In-wave reduction (deterministic fixed tree), one partial per block
  for (int off = 16; off > 0; off >>= 1)
    acc += __shfl_down(acc, off, 32);
  if (tid == 0) {
    const int bid = (blockIdx.z * gridDim.y + blockIdx.y) * gridDim.x + blockIdx.x;
    partials[bid] = acc;
  }
}

__global__ __launch_bounds__(256)
void ncc3d_reduce(const float* __restrict__ partials, int n, float* __restrict__ out)
{
  __shared__ float sm[256];
  float a = 0.0f;
  for (int i = threadIdx.x; i < n; i += 256) a += partials[i];  // fixed order
  sm[threadIdx.x] = a;
  __syncthreads();
  for (int off = 128; off > 0; off >>= 1) {
    if ((int)threadIdx.x < off) sm[threadIdx.x] += sm[threadIdx.x + off];
    __syncthreads();
  }
  if (threadIdx.x == 0)
    out[0] = -sm[0] / (float)((long)BD * DD * HD * WD);
}

extern "C" void kernel_launch(void* const* d_in, const int* in_sizes, int n_in,
                              void* d_out, int out_size, void* d_ws, size_t ws_size,
                              hipStream_t stream) {
  (void)in_sizes; (void)n_in; (void)out_size; (void)ws_size;
  const float* I = (const float*)d_in[0];
  const float* T = (const float*)d_in[1];
  float* out = (float*)d_out;
  float* partials = (float*)d_ws;

  const int tilesW = WD / 16;            // 10
  const int tilesH = HD / 16;            // 12
  dim3 grid(tilesW * tilesH, DD / CD, BD);   // 120 x 8 x 2 = 1920 blocks
  ncc3d_partial<<<grid, 32, 0, stream>>>(I, T, partials);

  const int nparts = tilesW * tilesH * (DD / CD) * BD;
  ncc3d_reduce<<<1, 256, 0, stream>>>(partials, nparts, out);
}